// UniPhyBlock_71708773974673
// MI455X (gfx1250) — compile-verified
//
#include <hip/hip_runtime.h>

// ---------------------------------------------------------------------------
// UniPhyBlock on gfx1250 (MI455X).  All channel-mixing einsums run on
// V_WMMA_F32_16X16X4_F32 (full-precision f32 WMMA, wave32).
// Shapes: B=2 T=12 D=64 H=W=48 -> NIMG=24 images, P=2304 pixels, M=55296 rows.
// Complex GEMMs are fused: one kernel produces both re/im outputs with
// multiple independent WMMA accumulator chains per wave.  Weight panels are
// staged k-pair-interleaved in LDS so each B fragment is a single aligned
// ds_load_b64; epilogue uses b128 stores.
// ---------------------------------------------------------------------------

typedef __attribute__((ext_vector_type(2))) float v2f;
typedef __attribute__((ext_vector_type(4))) float v4f;
typedef __attribute__((ext_vector_type(8))) float v8f;

#define B_    2
#define T_    12
#define D_    64
#define HH_   48
#define WW_   48
#define P_    2304            // HH_*WW_
#define NIMG  24              // B_*T_
#define MROWS 55296           // NIMG*P_
#define NOISE_SCALE 0.05f
#define LN_EPS 1e-5f

__device__ __forceinline__ float d_sigmoid(float v) { return 1.0f / (1.0f + __expf(-v)); }
__device__ __forceinline__ float d_silu(float v)    { return v * d_sigmoid(v); }

// ---------------------------------------------------------------------------
// Fused (optionally complex) WMMA GEMM over planar activations.
//
//   accR = A1 @ P0 + (DUAL) A2 @ P1
//   accI = A1 @ P2 + (DUAL) A2 @ P3          (only if COUT)
//   outR = act(accR + b1) + (HAS_RES) res1 ; outI likewise with b2/res2
//
// Activations are planar: A[(img*C + c)*P_ + p].  Weight panel k (o,c) =
// sgn * Wp[o*so + c*sc]; panels staged sign-folded into LDS in k-pair-
// interleaved order:  lds[(c>>1)*2*SPAN + col*2 + (c&1)], so the f32
// 16x16x4 B fragment (W(ka,col), W(ka+1,col)) is one aligned ds_load_b64.
//
// block = 256 threads = 8 waves; wave = one 16-row M tile; blockIdx.y = one
// SPAN-wide column panel (SPAN in {16,32,64}).  C is a template param so the
// K loop fully unrolls: straight-line clauses of loads + WMMAs with
// SPAN/16 * (1+COUT) independent accumulator chains per wave.
// ACT: 0 none, 1 silu, 2 sigmoid.
// ---------------------------------------------------------------------------
template <int ACT, bool DUAL, bool COUT, bool HAS_RES, int C, int SPAN>
__global__ __launch_bounds__(256) void k_cgemm(
    const float* __restrict__ A1, const float* __restrict__ A2,
    const float* __restrict__ Wp0, int so0, int sc0, float gn0,
    const float* __restrict__ Wp1, int so1, int sc1, float gn1,
    const float* __restrict__ Wp2, int so2, int sc2, float gn2,
    const float* __restrict__ Wp3, int so3, int sc3, float gn3,
    const float* __restrict__ b1, const float* __restrict__ b2,
    const float* __restrict__ res1, const float* __restrict__ res2,
    float* __restrict__ outR, float* __restrict__ outI, int O)
{
    constexpr int NP  = (DUAL ? 2 : 1) * (COUT ? 2 : 1);
    constexpr int CS  = C * SPAN;           // floats per panel
    constexpr int CP  = CS / 2;             // v2f per panel
    constexpr int OT  = SPAN / 16;          // 16-col tiles per wave
    constexpr int SL1 = DUAL ? 1 : 0;       // LDS slot of P1
    constexpr int SL2 = DUAL ? 2 : 1;       // LDS slot of P2
    __shared__ v2f lds2[NP * CP];
    float* ldsf = (float*)lds2;

    const int tid   = threadIdx.x;
    const int obase = blockIdx.y * SPAN;

    // Stage sign-folded weight panels, k-pair-interleaved.
    // (CS is a multiple of 256: no divergence.)
    for (int i = tid; i < CS; i += 256) {
        const int c  = i / SPAN;
        const int ol = i % SPAN;
        const int o  = obase + ol;
        const int fi = (c >> 1) * (2 * SPAN) + ol * 2 + (c & 1);
        ldsf[fi] = gn0 * Wp0[o * so0 + c * sc0];
        if constexpr (DUAL)         ldsf[SL1 * CS + fi] = gn1 * Wp1[o * so1 + c * sc1];
        if constexpr (COUT)         ldsf[SL2 * CS + fi] = gn2 * Wp2[o * so2 + c * sc2];
        if constexpr (DUAL && COUT) ldsf[3   * CS + fi] = gn3 * Wp3[o * so3 + c * sc3];
    }
    __syncthreads();

    const int wave = tid >> 5;
    const int lane = tid & 31;
    const int m0   = (blockIdx.x * 8 + wave) * 16;   // P_ % 16 == 0
    const int img  = m0 / P_;
    const int p0   = m0 % P_;
    const int sel  = (lane < 16) ? 0 : 2;            // ISA 16x4 f32 A layout
    const int psel = (lane < 16) ? 0 : 1;            // B k-pair select
    const int ml   = lane & 15;

    const float* a1base = A1 + (img * C) * P_ + p0 + ml;
    const float* a2base = DUAL ? (A2 + (img * C) * P_ + p0 + ml) : nullptr;
    __builtin_prefetch(a1base, 0, 1);
    if constexpr (DUAL) __builtin_prefetch(a2base, 0, 1);

    v8f accR[OT];
    v8f accI[COUT ? OT : 1];
#pragma unroll
    for (int j = 0; j < OT; ++j) {
        accR[j] = v8f{0.f, 0.f, 0.f, 0.f, 0.f, 0.f, 0.f, 0.f};
        if constexpr (COUT) accI[j] = v8f{0.f, 0.f, 0.f, 0.f, 0.f, 0.f, 0.f, 0.f};
    }

#pragma unroll
    for (int k0 = 0; k0 < C; k0 += 4) {
        const int ka = k0 + sel;
        const int pb = (k0 >> 1) + psel;             // k-pair index
        v2f a1;
        a1.x = a1base[ka * P_];
        a1.y = a1base[(ka + 1) * P_];
        v2f a2;
        if constexpr (DUAL) {
            a2.x = a2base[ka * P_];
            a2.y = a2base[(ka + 1) * P_];
        }
#pragma unroll
        for (int j = 0; j < OT; ++j) {
            const int col = j * 16 + ml;
            const v2f b = lds2[pb * SPAN + col];
            accR[j] = __builtin_amdgcn_wmma_f32_16x16x4_f32(false, a1, false, b,
                                                            (short)0, accR[j], false, false);
            if constexpr (DUAL) {
                const v2f bb = lds2[SL1 * CP + pb * SPAN + col];
                accR[j] = __builtin_amdgcn_wmma_f32_16x16x4_f32(false, a2, false, bb,
                                                                (short)0, accR[j], false, false);
            }
            if constexpr (COUT) {
                const v2f bi = lds2[SL2 * CP + pb * SPAN + col];
                accI[j] = __builtin_amdgcn_wmma_f32_16x16x4_f32(false, a1, false, bi,
                                                                (short)0, accI[j], false, false);
                if constexpr (DUAL) {
                    const v2f bj = lds2[3 * CP + pb * SPAN + col];
                    accI[j] = __builtin_amdgcn_wmma_f32_16x16x4_f32(false, a2, false, bj,
                                                                    (short)0, accI[j], false, false);
                }
            }
        }
    }

    // D layout: VGPR r holds (row = r + 8*(lane>=16), col = lane%16).
    // Rows 0..3 and 4..7 are contiguous pixels: two b128 stores per tile.
    const int prow0 = p0 + ((lane < 16) ? 0 : 8);
#pragma unroll
    for (int j = 0; j < OT; ++j) {
        const int   ocol = obase + j * 16 + ml;
        const float bvR  = b1 ? b1[ocol] : 0.0f;
        float* obR = outR + (img * O + ocol) * P_ + prow0;
        v4f rlo{}, rhi{};
        if constexpr (HAS_RES) {
            const float* rbR = res1 + (img * O + ocol) * P_ + prow0;
            rlo = *(const v4f*)rbR;
            rhi = *(const v4f*)(rbR + 4);
        }
        v4f vlo, vhi;
#pragma unroll
        for (int r = 0; r < 8; ++r) {
            float v = accR[j][r] + bvR;
            if constexpr (ACT == 1) v = d_silu(v);
            if constexpr (ACT == 2) v = d_sigmoid(v);
            if constexpr (HAS_RES)  v += (r < 4) ? rlo[r] : rhi[r - 4];
            if (r < 4) vlo[r] = v; else vhi[r - 4] = v;
        }
        *(v4f*)obR       = vlo;
        *(v4f*)(obR + 4) = vhi;
        if constexpr (COUT) {
            const float bvI = b2 ? b2[ocol] : 0.0f;
            float* obI = outI + (img * O + ocol) * P_ + prow0;
            v4f ilo{}, ihi{};
            if constexpr (HAS_RES) {
                const float* rbI = res2 + (img * O + ocol) * P_ + prow0;
                ilo = *(const v4f*)rbI;
                ihi = *(const v4f*)(rbI + 4);
            }
            v4f wlo, whi;
#pragma unroll
            for (int r = 0; r < 8; ++r) {
                float v = accI[j][r] + bvI;
                if constexpr (ACT == 1) v = d_silu(v);
                if constexpr (ACT == 2) v = d_sigmoid(v);
                if constexpr (HAS_RES)  v += (r < 4) ? ilo[r] : ihi[r - 4];
                if (r < 4) wlo[r] = v; else whi[r - 4] = v;
            }
            *(v4f*)obI       = wlo;
            *(v4f*)(obI + 4) = whi;
        }
    }
}

// ---------------------------------------------------------------------------
// Depthwise 3x3, SAME zero padding, optional fused SiLU.
// ---------------------------------------------------------------------------
template <bool SILU>
__global__ __launch_bounds__(256) void k_dwconv(
    const float* __restrict__ in, const float* __restrict__ kw,
    float* __restrict__ out)
{
    const int idx = blockIdx.x * 256 + threadIdx.x;   // NIMG*D_*P_ = 3538944
    if (idx >= NIMG * D_ * P_) return;
    const int p = idx % P_;
    const int c = (idx / P_) % D_;
    const int n = idx / (D_ * P_);
    const int hh = p / WW_, ww = p % WW_;
    const float* base = in + (n * D_ + c) * P_;
    const float* kk   = kw + c * 9;
    float s = 0.0f;
#pragma unroll
    for (int dy = -1; dy <= 1; ++dy)
#pragma unroll
        for (int dx = -1; dx <= 1; ++dx) {
            const int y = hh + dy, x = ww + dx;
            if (y >= 0 && y < HH_ && x >= 0 && x < WW_)
                s += kk[(dy + 1) * 3 + (dx + 1)] * base[y * WW_ + x];
        }
    if constexpr (SILU) s = d_silu(s);
    out[idx] = s;
}

// ---------------------------------------------------------------------------
// Per-(n,e) spatial reductions: sums of re, im, |z|, |z|^2 over 2304 pixels.
// stats[ne*4 + {0:|z|, 1:|z|^2, 2:re, 3:im}]
// ---------------------------------------------------------------------------
__global__ __launch_bounds__(256) void k_stats(
    const float* __restrict__ xe_re, const float* __restrict__ xe_im,
    float* __restrict__ stats)
{
    const int ne = blockIdx.x;                 // n*64 + e, 1536 blocks
    const float* re = xe_re + ne * P_;
    const float* im = xe_im + ne * P_;
    float sm = 0.f, sm2 = 0.f, sr = 0.f, si = 0.f;
    for (int p = threadIdx.x; p < P_; p += 256) {
        const float r = re[p], i = im[p];
        const float m = sqrtf(r * r + i * i);
        sm += m; sm2 += m * m; sr += r; si += i;
    }
    __shared__ float sh[4 * 256];
    const int tid = threadIdx.x;
    sh[tid] = sm; sh[256 + tid] = sm2; sh[512 + tid] = sr; sh[768 + tid] = si;
    __syncthreads();
    for (int off = 128; off > 0; off >>= 1) {
        if (tid < off) {
            sh[tid]       += sh[tid + off];
            sh[256 + tid] += sh[256 + tid + off];
            sh[512 + tid] += sh[512 + tid + off];
            sh[768 + tid] += sh[768 + tid + off];
        }
        __syncthreads();
    }
    if (tid == 0) {
        stats[ne * 4 + 0] = sh[0];
        stats[ne * 4 + 1] = sh[256];
        stats[ne * 4 + 2] = sh[512];
        stats[ne * 4 + 3] = sh[768];
    }
}

// ---------------------------------------------------------------------------
// Flux recurrence over T (sequential, per (b,e)) + gate + exp-operator coeffs.
// 1 block, 128 threads.  Writes flux_out (interleaved complex) to d_out seg 2.
// ---------------------------------------------------------------------------
__global__ __launch_bounds__(128) void k_flux(
    const float* __restrict__ stats, const float* __restrict__ dt_seq,
    const float* __restrict__ fp_re, const float* __restrict__ fp_im,
    const float* __restrict__ flux_log_decay,
    const float* __restrict__ src_sr, const float* __restrict__ src_si,
    const float* __restrict__ gate_w, const float* __restrict__ gate_b,
    const float* __restrict__ pw1, const float* __restrict__ pb1,
    const float* __restrict__ pw2, const float* __restrict__ pb2,
    const float* __restrict__ log_decay, const float* __restrict__ freq,
    float* __restrict__ src_re, float* __restrict__ src_im,
    float* __restrict__ gate_s,
    float* __restrict__ opd_re, float* __restrict__ opd_im,
    float* __restrict__ opf_re, float* __restrict__ opf_im,
    float* __restrict__ out_flux)
{
    const int tid = threadIdx.x;       // b*64 + e
    if (tid >= B_ * D_) return;
    const int b = tid >> 6, e = tid & 63;
    const float fld = flux_log_decay[e];
    const float ssr = src_sr[e], ssi = src_si[e];
    const float gw = gate_w[e], gb = gate_b[e];
    const float lre = -expf(log_decay[e]);
    const float lim = freq[e];
    const float fpr = fp_re[b * D_ + e], fpi = fp_im[b * D_ + e];

    float fre = 0.f, fim = 0.f, cum = 1.f;
    for (int t = 0; t < T_; ++t) {
        const int s  = (b * T_ + t) * D_ + e;
        const float dt = dt_seq[b * T_ + t];
        const float inv = 1.0f / (float)P_;
        const float mag_mean = stats[s * 4 + 0] * inv;
        const float var  = fmaxf(stats[s * 4 + 1] * inv - mag_mean * mag_mean, 0.f);
        const float mag_std = sqrtf(var);
        const float mre = stats[s * 4 + 2] * inv;
        const float mim = stats[s * 4 + 3] * inv;
        // pooled-scale MLP: 2 -> 4 (silu) -> 1 -> tanh
        float acc = pb2[0];
#pragma unroll
        for (int j = 0; j < 4; ++j) {
            const float h = d_silu(mag_mean * pw1[j] + mag_std * pw1[4 + j] + pb1[j]);
            acc += h * pw2[j];
        }
        const float scale = tanhf(acc);
        const float xmr = mre * (1.0f + scale), xmi = mim * (1.0f + scale);
        // flux scan: h_t = a_t h_{t-1} + x_t
        const float af = expf(-expf(fld) * dt);
        fre = af * fre + xmr * dt;
        fim = af * fim + xmi * dt;
        cum *= af;
        const float Fr = fre + fpr * cum;
        const float Fi = fim + fpi * cum;
        src_re[s] = Fr * ssr - Fi * ssi;
        src_im[s] = Fr * ssi + Fi * ssr;
        gate_s[s] = d_sigmoid(sqrtf(Fr * Fr + Fi * Fi) * gw + gb);
        // op_decay = exp(lam*dt);  op_forcing = (op_decay-1)/lam
        const float er  = expf(lre * dt);
        const float odr = er * cosf(lim * dt);
        const float odi = er * sinf(lim * dt);
        opd_re[s] = odr; opd_im[s] = odi;
        const float nr = odr - 1.0f, ni = odi;
        const float den = lre * lre + lim * lim;
        opf_re[s] = (nr * lre + ni * lim) / den;
        opf_im[s] = (ni * lre - nr * lim) / den;
        if (t == T_ - 1) {
            out_flux[(b * D_ + e) * 2 + 0] = Fr;
            out_flux[(b * D_ + e) * 2 + 1] = Fi;
        }
    }
}

// ---------------------------------------------------------------------------
// Forcing: u_t = (xe*gc + src*(1-gc)) * op_forcing + noise (+ h_prev@t0)
// ---------------------------------------------------------------------------
__global__ __launch_bounds__(256) void k_forcing(
    const float* __restrict__ xe_re, const float* __restrict__ xe_im,
    const float* __restrict__ sgate,
    const float* __restrict__ gate_s,
    const float* __restrict__ src_re, const float* __restrict__ src_im,
    const float* __restrict__ opf_re, const float* __restrict__ opf_im,
    const float* __restrict__ opd_re, const float* __restrict__ opd_im,
    const float* __restrict__ noise, const float* __restrict__ dt_seq,
    const float* __restrict__ hp_re, const float* __restrict__ hp_im,
    float* __restrict__ ut_re, float* __restrict__ ut_im)
{
    const int idx = blockIdx.x * 256 + threadIdx.x;     // (n*64+e)*P_ + p
    if (idx >= NIMG * D_ * P_) return;
    const int p = idx % P_;
    const int e = (idx / P_) % D_;
    const int n = idx / (D_ * P_);
    const int b = n / T_, t = n % T_;
    const int s = n * D_ + e;

    const float gc = gate_s[s] * sgate[idx];
    const float fr = xe_re[idx] * gc + src_re[s] * (1.0f - gc);
    const float fi = xe_im[idx] * gc + src_im[s] * (1.0f - gc);
    float ur = fr * opf_re[s] - fi * opf_im[s];
    float ui = fr * opf_im[s] + fi * opf_re[s];
    // noise (real) -- layout (b,t,h,w,d)
    const float nv = NOISE_SCALE * sqrtf(dt_seq[n]);
    ur += nv * noise[n * (P_ * D_) + p * D_ + e];
    if (t == 0) {
        const float hr = hp_re[(b * P_ + p) * D_ + e];
        const float hi = hp_im[(b * P_ + p) * D_ + e];
        const float odr = opd_re[(b * T_) * D_ + e];
        const float odi = opd_im[(b * T_) * D_ + e];
        ur += hr * odr - hi * odi;
        ui += hr * odi + hi * odr;
    }
    ut_re[idx] = ur;
    ut_im[idx] = ui;
}

// ---------------------------------------------------------------------------
// Temporal scan over T (sequential per (b,e,p)); also writes h_out (seg 1).
// ---------------------------------------------------------------------------
__global__ __launch_bounds__(256) void k_tscan(
    const float* __restrict__ ut_re, const float* __restrict__ ut_im,
    const float* __restrict__ opd_re, const float* __restrict__ opd_im,
    float* __restrict__ uo_re, float* __restrict__ uo_im,
    float* __restrict__ out_h)
{
    const int idx = blockIdx.x * 256 + threadIdx.x;     // b*64*P_ + e*P_ + p
    if (idx >= B_ * D_ * P_) return;
    const int p = idx % P_;
    const int e = (idx / P_) % D_;
    const int b = idx / (D_ * P_);
    float ur = 0.f, ui = 0.f;
    for (int t = 0; t < T_; ++t) {
        const int s = (b * T_ + t) * D_ + e;
        const int q = s * P_ + p;
        const float ar = opd_re[s], ai = opd_im[s];
        const float nr = ar * ur - ai * ui + ut_re[q];
        const float ni = ar * ui + ai * ur + ut_im[q];
        ur = nr; ui = ni;
        uo_re[q] = ur;
        uo_im[q] = ui;
    }
    out_h[((b * P_ + p) * D_ + e) * 2 + 0] = ur;
    out_h[((b * P_ + p) * D_ + e) * 2 + 1] = ui;
}

// ---------------------------------------------------------------------------
// Complex 64x64 Gauss-Jordan inverse of Wb (single block, augmented in ws).
// ---------------------------------------------------------------------------
__global__ __launch_bounds__(256) void k_cinv(
    const float* __restrict__ wre, const float* __restrict__ wim,
    float* __restrict__ ar, float* __restrict__ ai,
    float* __restrict__ invr, float* __restrict__ invi)
{
    const int tid = threadIdx.x;
    for (int i = tid; i < 64 * 128; i += 256) {
        const int r = i >> 7, c = i & 127;
        if (c < 64) { ar[i] = wre[r * 64 + c]; ai[i] = wim[r * 64 + c]; }
        else        { ar[i] = (c - 64 == r) ? 1.0f : 0.0f; ai[i] = 0.0f; }
    }
    __syncthreads();
    __shared__ int   s_piv;
    __shared__ float s_pr, s_pi;
    __shared__ float fcr[64], fci[64];
    for (int k = 0; k < 64; ++k) {
        if (tid == 0) {
            int piv = k; float best = -1.0f;
            for (int r = k; r < 64; ++r) {
                const float m = ar[r * 128 + k] * ar[r * 128 + k] +
                                ai[r * 128 + k] * ai[r * 128 + k];
                if (m > best) { best = m; piv = r; }
            }
            s_piv = piv;
        }
        __syncthreads();
        const int piv = s_piv;
        if (piv != k) {
            for (int c = tid; c < 128; c += 256) {
                const float tr = ar[k * 128 + c], ti = ai[k * 128 + c];
                ar[k * 128 + c] = ar[piv * 128 + c];
                ai[k * 128 + c] = ai[piv * 128 + c];
                ar[piv * 128 + c] = tr;
                ai[piv * 128 + c] = ti;
            }
        }
        __syncthreads();
        if (tid == 0) {
            const float pr = ar[k * 128 + k], pi = ai[k * 128 + k];
            const float d = pr * pr + pi * pi;
            s_pr = pr / d; s_pi = -pi / d;
        }
        __syncthreads();
        for (int c = tid; c < 128; c += 256) {
            const float vr = ar[k * 128 + c], vi = ai[k * 128 + c];
            ar[k * 128 + c] = vr * s_pr - vi * s_pi;
            ai[k * 128 + c] = vr * s_pi + vi * s_pr;
        }
        __syncthreads();
        for (int r = tid; r < 64; r += 256) { fcr[r] = ar[r * 128 + k]; fci[r] = ai[r * 128 + k]; }
        __syncthreads();
        for (int i = tid; i < 64 * 128; i += 256) {
            const int r = i >> 7, c = i & 127;
            if (r == k) continue;
            const float kr = ar[k * 128 + c], ki = ai[k * 128 + c];
            ar[i] -= fcr[r] * kr - fci[r] * ki;
            ai[i] -= fcr[r] * ki + fci[r] * kr;
        }
        __syncthreads();
    }
    for (int i = tid; i < 64 * 64; i += 256) {
        const int r = i >> 6, c = i & 63;
        invr[i] = ar[r * 128 + 64 + c];
        invi[i] = ai[r * 128 + 64 + c];
    }
}

// ---------------------------------------------------------------------------
// LayerNorm over 128 channels ([dec_re(64); dec_im(64)]) per pixel.
// ---------------------------------------------------------------------------
__global__ __launch_bounds__(256) void k_ln(
    const float* __restrict__ dre, const float* __restrict__ dim,
    const float* __restrict__ gamma, const float* __restrict__ beta,
    float* __restrict__ xn_re, float* __restrict__ xn_im)
{
    const int idx = blockIdx.x * 256 + threadIdx.x;     // n*P_ + p
    if (idx >= NIMG * P_) return;
    const int n = idx / P_, p = idx % P_;
    float s = 0.f, s2 = 0.f;
    for (int d = 0; d < D_; ++d) {
        const float vr = dre[(n * D_ + d) * P_ + p];
        const float vi = dim[(n * D_ + d) * P_ + p];
        s += vr + vi; s2 += vr * vr + vi * vi;
    }
    const float mu = s * (1.0f / 128.0f);
    const float var = s2 * (1.0f / 128.0f) - mu * mu;
    const float rs = rsqrtf(var + LN_EPS);
    for (int d = 0; d < D_; ++d) {
        const int q = (n * D_ + d) * P_ + p;
        xn_re[q] = (dre[q] - mu) * rs * gamma[d] + beta[d];
        xn_im[q] = (dim[q] - mu) * rs * gamma[D_ + d] + beta[D_ + d];
    }
}

// ---------------------------------------------------------------------------
// Final combine: where(|dt|<=1e-12, complex(xmix,0), dres) -> interleaved out.
// ---------------------------------------------------------------------------
__global__ __launch_bounds__(256) void k_combine(
    const float* __restrict__ xmix,
    const float* __restrict__ dres_re, const float* __restrict__ dres_im,
    const float* __restrict__ dt_seq, float* __restrict__ out)
{
    const int idx = blockIdx.x * 256 + threadIdx.x;     // (n*64+e)*P_ + p
    if (idx >= NIMG * D_ * P_) return;
    const int n = idx / (D_ * P_);
    const bool zm = fabsf(dt_seq[n]) <= 1e-12f;
    out[idx * 2 + 0] = zm ? xmix[idx] : dres_re[idx];
    out[idx * 2 + 1] = zm ? 0.0f      : dres_im[idx];
}

// ---------------------------------------------------------------------------
// Host-side orchestration.
// ---------------------------------------------------------------------------
extern "C" void kernel_launch(void* const* d_in, const int* in_sizes, int n_in,
                              void* d_out, int out_size, void* d_ws, size_t ws_size,
                              hipStream_t stream)
{
    const float* x       = (const float*)d_in[0];
    const float* hp_re   = (const float*)d_in[1];
    const float* hp_im   = (const float*)d_in[2];
    const float* dt_seq  = (const float*)d_in[3];
    const float* fp_re   = (const float*)d_in[4];
    const float* fp_im   = (const float*)d_in[5];
    const float* noise   = (const float*)d_in[6];
    const float* mixer_dw   = (const float*)d_in[7];
    const float* mixer_pw_w = (const float*)d_in[8];
    const float* mixer_pw_b = (const float*)d_in[9];
    const float* ln_gamma   = (const float*)d_in[10];
    const float* ln_beta    = (const float*)d_in[11];
    const float* basis_w_re = (const float*)d_in[12];
    const float* basis_w_im = (const float*)d_in[13];
    const float* log_decay  = (const float*)d_in[14];
    const float* freq       = (const float*)d_in[15];
    const float* flux_ld    = (const float*)d_in[16];
    const float* src_sr     = (const float*)d_in[17];
    const float* src_si     = (const float*)d_in[18];
    const float* gate_w     = (const float*)d_in[19];
    const float* gate_b     = (const float*)d_in[20];
    const float* pool_w1    = (const float*)d_in[21];
    const float* pool_b1    = (const float*)d_in[22];
    const float* pool_w2    = (const float*)d_in[23];
    const float* pool_b2    = (const float*)d_in[24];
    const float* sg_w1      = (const float*)d_in[25];
    const float* sg_b1      = (const float*)d_in[26];
    const float* sg_dw      = (const float*)d_in[27];
    const float* sg_w2      = (const float*)d_in[28];
    const float* sg_b2      = (const float*)d_in[29];
    const float* ffn_w1_re  = (const float*)d_in[30];
    const float* ffn_w1_im  = (const float*)d_in[31];
    const float* ffn_b1_re  = (const float*)d_in[32];
    const float* ffn_b1_im  = (const float*)d_in[33];
    const float* ffn_w2_re  = (const float*)d_in[34];
    const float* ffn_w2_im  = (const float*)d_in[35];
    const float* ffn_b2_re  = (const float*)d_in[36];
    const float* ffn_b2_im  = (const float*)d_in[37];

    float* ws = (float*)d_ws;
    constexpr size_t S = (size_t)NIMG * D_ * P_;   // 3,538,944 floats
    float* s0 = ws + 0 * S;       // dwconv out / g1 / sgate scratch
    float* s1 = ws + 1 * S;       // xmix (lives until final combine)
    float* s2 = ws + 2 * S;       // xe_re -> uo_re -> xn_re -> dres_re
    float* s3 = ws + 3 * S;       // xe_im -> uo_im -> xn_im -> dres_im
    float* s4 = ws + 4 * S;       // g1conv -> ut_re -> dec_re
    float* s5 = ws + 5 * S;       // ut_im -> dec_im
    float* h0 = ws + 6 * S;       // hm_re (2S)
    float* h1 = ws + 8 * S;       // hm_im (2S)
    float* sm = ws + 10 * S;      // small region
    float* st_stats = sm;                    // 1536*4
    float* src_re   = sm + 6144;             // 1536 each below
    float* src_im   = sm + 7680;
    float* gate_s   = sm + 9216;
    float* opd_re   = sm + 10752;
    float* opd_im   = sm + 12288;
    float* opf_re   = sm + 13824;
    float* opf_im   = sm + 15360;
    float* augr     = sm + 16896;            // 8192
    float* augi     = sm + 25088;            // 8192
    float* winv_re  = sm + 33280;            // 4096
    float* winv_im  = sm + 37376;            // 4096

    float* out0 = (float*)d_out;                 // combined, interleaved complex
    float* out1 = out0 + (size_t)2 * S;          // h_out
    float* out2 = out1 + (size_t)2 * B_ * P_ * D_; // flux_out

    const dim3 blk(256);
    const dim3 gElem((NIMG * D_ * P_ + 255) / 256);
    const int  GX = MROWS / 128;                 // 432

    // 1) mixer: y = silu(dwconv(x)); xmix = x + (y @ pw^T + b)
    k_dwconv<true><<<gElem, blk, 0, stream>>>(x, mixer_dw, s0);
    k_cgemm<0, false, false, true, 64, 64><<<dim3(GX, 1), blk, 0, stream>>>(
        s0, nullptr,
        mixer_pw_w, 64, 1, 1.f,  nullptr, 0, 0, 0.f,
        nullptr, 0, 0, 0.f,      nullptr, 0, 0, 0.f,
        mixer_pw_b, nullptr, x, nullptr, s1, nullptr, 64);

    // 2) eigen-basis (fused re+im): xe = xmix @ (Wre + i Wim)
    k_cgemm<0, false, true, false, 64, 32><<<dim3(GX, 2), blk, 0, stream>>>(
        s1, nullptr,
        basis_w_re, 1, 64, 1.f,  nullptr, 0, 0, 0.f,
        basis_w_im, 1, 64, 1.f,  nullptr, 0, 0, 0.f,
        nullptr, nullptr, nullptr, nullptr, s2, s3, 64);

    // 3) pooled stats + flux scan + operator coefficients
    k_stats<<<dim3(NIMG * D_), blk, 0, stream>>>(s2, s3, st_stats);
    k_flux<<<dim3(1), dim3(128), 0, stream>>>(
        st_stats, dt_seq, fp_re, fp_im, flux_ld, src_sr, src_si, gate_w, gate_b,
        pool_w1, pool_b1, pool_w2, pool_b2, log_decay, freq,
        src_re, src_im, gate_s, opd_re, opd_im, opf_re, opf_im, out2);

    // 4) spatial gate: sigmoid(dwconv(silu([re;im] @ sg_w1^T + b1)) @ sg_w2^T + b2)
    k_cgemm<1, true, false, false, 64, 32><<<dim3(GX, 2), blk, 0, stream>>>(
        s2, s3,
        sg_w1, 128, 1, 1.f,      sg_w1 + 64, 128, 1, 1.f,
        nullptr, 0, 0, 0.f,      nullptr, 0, 0, 0.f,
        sg_b1, nullptr, nullptr, nullptr, s0, nullptr, 64);
    k_dwconv<false><<<gElem, blk, 0, stream>>>(s0, sg_dw, s4);
    k_cgemm<2, false, false, false, 64, 64><<<dim3(GX, 1), blk, 0, stream>>>(
        s4, nullptr,
        sg_w2, 64, 1, 1.f,       nullptr, 0, 0, 0.f,
        nullptr, 0, 0, 0.f,      nullptr, 0, 0, 0.f,
        sg_b2, nullptr, nullptr, nullptr, s0, nullptr, 64);

    // 5) forcing + temporal scan (writes h_out)
    k_forcing<<<gElem, blk, 0, stream>>>(
        s2, s3, s0, gate_s, src_re, src_im, opf_re, opf_im, opd_re, opd_im,
        noise, dt_seq, hp_re, hp_im, s4, s5);
    k_tscan<<<dim3((B_ * D_ * P_) / 256), blk, 0, stream>>>(
        s4, s5, opd_re, opd_im, s2, s3, out1);

    // 6) decode (fused complex GEMM): dec = u_out @ Wb_inv
    k_cinv<<<dim3(1), blk, 0, stream>>>(basis_w_re, basis_w_im, augr, augi,
                                        winv_re, winv_im);
    k_cgemm<0, true, true, false, 64, 32><<<dim3(GX, 2), blk, 0, stream>>>(
        s2, s3,
        winv_re, 1, 64, 1.f,     winv_im, 1, 64, -1.f,
        winv_im, 1, 64, 1.f,     winv_re, 1, 64, 1.f,
        nullptr, nullptr, nullptr, nullptr, s4, s5, 64);

    // 7) LayerNorm over 128 channels
    k_ln<<<dim3((NIMG * P_) / 256), blk, 0, stream>>>(
        s4, s5, ln_gamma, ln_beta, s2, s3);

    // 8) complex FFN: hm = silu_c(xn @ w1 + b1); dres = dec + hm @ w2 + b2
    k_cgemm<1, true, true, false, 64, 32><<<dim3(GX, 4), blk, 0, stream>>>(
        s2, s3,
        ffn_w1_re, 1, 128, 1.f,  ffn_w1_im, 1, 128, -1.f,
        ffn_w1_im, 1, 128, 1.f,  ffn_w1_re, 1, 128, 1.f,
        ffn_b1_re, ffn_b1_im, nullptr, nullptr, h0, h1, 128);
    k_cgemm<0, true, true, true, 128, 16><<<dim3(GX, 4), blk, 0, stream>>>(
        h0, h1,
        ffn_w2_re, 1, 64, 1.f,   ffn_w2_im, 1, 64, -1.f,
        ffn_w2_im, 1, 64, 1.f,   ffn_w2_re, 1, 64, 1.f,
        ffn_b2_re, ffn_b2_im, s4, s5, s2, s3, 64);

    // 9) zero-dt passthrough + interleaved complex output
    k_combine<<<gElem, blk, 0, stream>>>(s1, s2, s3, dt_seq, out0);

    (void)in_sizes; (void)n_in; (void)out_size; (void)ws_size;
}